// P2M_65189013618886
// MI455X (gfx1250) — compile-verified
//
#include <hip/hip_runtime.h>
#include <hip/hip_bf16.h>
#include <stdint.h>

#ifndef __has_builtin
#define __has_builtin(x) 0
#endif
#if __has_builtin(__builtin_amdgcn_global_load_async_to_lds_b32)
#define HAVE_ASYNC_LDS 1
#endif
#if __has_builtin(__builtin_amdgcn_s_wait_asynccnt)
#define HAVE_WAIT_ASYNC 1
#endif

typedef __attribute__((ext_vector_type(2))) float v2f;
typedef __attribute__((ext_vector_type(8))) float v8f;

#define NCOL 3843           // 3*1280 + 3
#define CPOOL 1280
#define PAD_KEY 0xFFFFFFFFu

// ---------------- edge keys: (min<<16)|max for edges (i1,i2) then (i2,i3) ----------------
__global__ void k_build_keys(const int* __restrict__ faces, int F, unsigned* __restrict__ keys)
{
    int i = blockIdx.x * blockDim.x + threadIdx.x;
    if (i >= 2 * F) return;
    int f = (i < F) ? i : (i - F);
    int a, b;
    if (i < F) { a = faces[3*f + 0]; b = faces[3*f + 1]; }
    else       { a = faces[3*f + 1]; b = faces[3*f + 2]; }
    int lo = a < b ? a : b;
    int hi = a < b ? b : a;
    keys[i] = ((unsigned)lo << 16) | (unsigned)(hi & 0xFFFF);
}

// ---------------- single-workgroup LDS bitonic sort + unique (CDNA5 big-LDS path) --------
// Loads keys into LDS with gfx1250 async global->LDS copies, sorts P (pow2) entries,
// emits sorted unique keys padded with 0 (replicates jnp.unique fill_value=0), and U.
__global__ void k_sort_unique(const unsigned* __restrict__ keys, int count, int P,
                              unsigned* __restrict__ uniq, int* __restrict__ Uout)
{
    extern __shared__ unsigned smem[];
    unsigned* s    = smem;          // P entries
    unsigned* csum = smem + P;      // blockDim.x entries
    __shared__ int totalU;
    const int tid = threadIdx.x;
    const int nt  = blockDim.x;

    for (int i = tid; i < count; i += nt) {
#ifdef HAVE_ASYNC_LDS
        __builtin_amdgcn_global_load_async_to_lds_b32(
            (__attribute__((address_space(1))) int*)(keys + i),
            (__attribute__((address_space(3))) int*)(s + i), 0, 0);
#else
        s[i] = keys[i];
#endif
    }
    for (int i = count + tid; i < P; i += nt) s[i] = PAD_KEY;
#ifdef HAVE_ASYNC_LDS
#ifdef HAVE_WAIT_ASYNC
    __builtin_amdgcn_s_wait_asynccnt(0);
#else
    asm volatile("s_wait_asynccnt 0" ::: "memory");
#endif
#endif
    __syncthreads();

    // bitonic sort ascending
    for (int k = 2; k <= P; k <<= 1) {
        for (int j = k >> 1; j > 0; j >>= 1) {
            for (int i = tid; i < P; i += nt) {
                int l = i ^ j;
                if (l > i) {
                    unsigned a = s[i], b = s[l];
                    bool up = ((i & k) == 0);
                    bool sw = up ? (a > b) : (a < b);
                    if (sw) { s[i] = b; s[l] = a; }
                }
            }
            __syncthreads();
        }
    }

    // head-flag count per contiguous chunk, serial scan, emit uniques
    const int chunk = P / nt;
    const int base  = tid * chunk;
    int cnt = 0;
    for (int t = 0; t < chunk; ++t) {
        int i = base + t;
        unsigned vk = s[i];
        bool head = (vk != PAD_KEY) && (i == 0 || s[i - 1] != vk);
        if (head) cnt++;
    }
    csum[tid] = (unsigned)cnt;
    __syncthreads();
    if (tid == 0) {
        int run = 0;
        for (int t = 0; t < nt; ++t) { int c = (int)csum[t]; csum[t] = (unsigned)run; run += c; }
        totalU = run;
        *Uout  = run;
    }
    __syncthreads();
    int r = (int)csum[tid];
    for (int t = 0; t < chunk; ++t) {
        int i = base + t;
        unsigned vk = s[i];
        bool head = (vk != PAD_KEY) && (i == 0 || s[i - 1] != vk);
        if (head) uniq[r++] = vk;
    }
    for (int i = totalU + tid; i < count; i += nt) uniq[i] = 0u;  // jnp.unique fill_value=0
}

// ---------------- inverse: rank of each edge key in sorted unique list -------------------
__global__ void k_inverse(const unsigned* __restrict__ keys, int n,
                          const unsigned* __restrict__ uniq, const int* __restrict__ Uptr,
                          int* __restrict__ inv)
{
    int i = blockIdx.x * blockDim.x + threadIdx.x;
    if (i >= n) return;
    int U = *Uptr;
    unsigned key = keys[i];
    int lo = 0, hi = U;
    while (lo < hi) { int mid = (lo + hi) >> 1; if (uniq[mid] < key) lo = mid + 1; else hi = mid; }
    inv[i] = lo;
}

// ---------------- verts' = [verts ; 0.5*(verts[a]+verts[b])] -----------------------------
__global__ void k_midverts(const float* __restrict__ src, int Nsrc,
                           const unsigned* __restrict__ uniq, int nMid,
                           float* __restrict__ dst)
{
    int v = blockIdx.x * blockDim.x + threadIdx.x;
    if (v >= Nsrc + nMid) return;
    if (v < Nsrc) {
        dst[3*v+0] = src[3*v+0]; dst[3*v+1] = src[3*v+1]; dst[3*v+2] = src[3*v+2];
    } else {
        unsigned key = uniq[v - Nsrc];
        int a = key >> 16, b = key & 0xFFFF;
        dst[3*v+0] = 0.5f * (src[3*a+0] + src[3*b+0]);
        dst[3*v+1] = 0.5f * (src[3*a+1] + src[3*b+1]);
        dst[3*v+2] = 0.5f * (src[3*a+2] + src[3*b+2]);
    }
}

// ---------------- subdivided faces (order matches reference concat) ----------------------
__global__ void k_faces1(const int* __restrict__ faces0, int F0, const int* __restrict__ inv,
                         int N0, int* __restrict__ faces1)
{
    int f = blockIdx.x * blockDim.x + threadIdx.x;
    if (f >= F0) return;
    int i1 = faces0[3*f+0], i2 = faces0[3*f+1], i3 = faces0[3*f+2];
    int i4 = N0 + inv[f];
    int i5 = N0 + inv[F0 + f];
    int i6 = i5;
    faces1[3*f+0] = i1;            faces1[3*f+1] = i4;            faces1[3*f+2] = i6;
    int o = F0 + f;
    faces1[3*o+0] = i2;            faces1[3*o+1] = i4;            faces1[3*o+2] = i5;
    o = 2*F0 + f;
    faces1[3*o+0] = i3;            faces1[3*o+1] = i5;            faces1[3*o+2] = i6;
    o = 3*F0 + f;
    faces1[3*o+0] = i5;            faces1[3*o+1] = i4;            faces1[3*o+2] = i6;
}

// ---------------- projection + bilinear pooling over feat3/feat4/feat5 -------------------
__global__ void k_pool(const float* __restrict__ verts,
                       const float* __restrict__ f3, const float* __restrict__ f4,
                       const float* __restrict__ f5,
                       const float* __restrict__ cc, const float* __restrict__ cf,
                       const int* __restrict__ ims,
                       float* __restrict__ out, int colBase)
{
    int v = blockIdx.x;
    float vx = verts[3*v+0], vy = verts[3*v+1], vz = verts[3*v+2];
    float z  = vz + 0.8f;
    float px = cf[0] * vx / z + cc[0];
    float py = cf[1] * vy / z + cc[1];
    float imf = (float)(*ims);

    for (int lvl = 0; lvl < 3; ++lvl) {
        const float* fp; int H, C, cbase;
        if (lvl == 0)      { fp = f3; H = 56; C = 256; cbase = 0;   }
        else if (lvl == 1) { fp = f4; H = 28; C = 512; cbase = 256; }
        else               { fp = f5; H = 14; C = 512; cbase = 768; }
        float hf = (float)H;
        float xs = px * hf / imf;
        float ys = py * hf / imf;
        float x1 = floorf(xs), y1 = floorf(ys);
        float x2 = x1 + 1.0f,  y2 = y1 + 1.0f;
        int xi1 = (int)x1, yi1 = (int)y1, xi2 = (int)x2, yi2 = (int)y2;
        xi1 = xi1 < 0 ? 0 : (xi1 > H-1 ? H-1 : xi1);
        xi2 = xi2 < 0 ? 0 : (xi2 > H-1 ? H-1 : xi2);
        yi1 = yi1 < 0 ? 0 : (yi1 > H-1 ? H-1 : yi1);
        yi2 = yi2 < 0 ? 0 : (yi2 > H-1 ? H-1 : yi2);
        float w11 = (x2 - xs) * (y2 - ys);
        float w12 = (x2 - xs) * (ys - y1);
        float w21 = (xs - x1) * (y2 - ys);
        float w22 = (xs - x1) * (ys - y1);
        int HW = H * H;
        int o11 = xi1*H + yi1, o12 = xi1*H + yi2, o21 = xi2*H + yi1, o22 = xi2*H + yi2;
        for (int c = threadIdx.x; c < C; c += blockDim.x) {
            const float* pl = fp + (size_t)c * HW;
            float val = w11*pl[o11] + w12*pl[o12] + w21*pl[o21] + w22*pl[o22];
            out[(size_t)v * NCOL + colBase + cbase + c] = val;
        }
    }
}

// ---------------- WMMA midpoint propagation: D(16x16) = S(16x32) * X(32x16) -------------
// S has 0.5 at the two parent columns of each midpoint row; chained as 8 K=4 f32 WMMAs.
// One wave32 per 16-row x 16-channel tile; EXEC all ones, uniform control flow.
__global__ void k_prop_wmma(const unsigned* __restrict__ uniq, int totalTiles,
                            float* __restrict__ out, int col, int vBase)
{
    int wave = blockIdx.x * (blockDim.x >> 5) + (threadIdx.x >> 5);
    if (wave >= totalTiles) return;                 // wave-uniform guard
    int lane  = threadIdx.x & 31;
    int mlane = lane & 15;
    int isHi  = lane >> 4;
    int rb = wave / (CPOOL / 16);
    int cb = wave % (CPOOL / 16);
    int u0 = rb * 16;
    int cBase = col + cb * 16;

    v8f acc = {0.f, 0.f, 0.f, 0.f, 0.f, 0.f, 0.f, 0.f};
#pragma unroll
    for (int j = 0; j < 8; ++j) {
        // K-slice j covers global k = 4j..4j+3; this lane supplies k0=4j+2*isHi, k0+1.
        unsigned key = uniq[u0 + 2*j + isHi];       // edge for midpoint row m = 2j+isHi
        int a = key >> 16, b = key & 0xFFFF;
        float xa = out[(size_t)a * NCOL + cBase + mlane];
        float xb = out[(size_t)b * NCOL + cBase + mlane];
        float w  = (mlane == (2*j + isHi)) ? 0.5f : 0.0f;   // S'[mlane][k0] == S'[mlane][k0+1]
        v2f A = {w, w};
        v2f B = {xa, xb};
        acc = __builtin_amdgcn_wmma_f32_16x16x4_f32(false, A, false, B,
                                                    (short)0, acc, false, false);
    }
#pragma unroll
    for (int r = 0; r < 8; ++r) {
        int m = r + 8 * isHi;                       // C/D layout: VGPR r -> M = r (+8 hi half)
        out[(size_t)(vBase + u0 + m) * NCOL + cBase + mlane] = acc[r];
    }
}

// scalar tail for non-multiple-of-16 midpoint counts (not hit for the fixed sizes)
__global__ void k_prop_tail(const unsigned* __restrict__ uniq, int startMid, int nMid,
                            float* __restrict__ out, int col, int vBase)
{
    long idx = (long)blockIdx.x * blockDim.x + threadIdx.x;
    long total = (long)nMid * CPOOL;
    if (idx >= total) return;
    int m = (int)(idx / CPOOL), c = (int)(idx % CPOOL);
    unsigned key = uniq[startMid + m];
    int a = key >> 16, b = key & 0xFFFF;
    out[(size_t)(vBase + startMid + m) * NCOL + col + c] =
        0.5f * (out[(size_t)a * NCOL + col + c] + out[(size_t)b * NCOL + col + c]);
}

// ---------------- final 3 vertex-coordinate columns --------------------------------------
__global__ void k_vertcols(const float* __restrict__ verts2, int N2, float* __restrict__ out)
{
    int v = blockIdx.x * blockDim.x + threadIdx.x;
    if (v >= N2) return;
    out[(size_t)v * NCOL + 3840 + 0] = verts2[3*v+0];
    out[(size_t)v * NCOL + 3840 + 1] = verts2[3*v+1];
    out[(size_t)v * NCOL + 3840 + 2] = verts2[3*v+2];
}

static inline int ceil_div_i(int a, int b) { return (a + b - 1) / b; }
static inline int npow2_i(int x) { int p = 1; while (p < x) p <<= 1; return p; }

extern "C" void kernel_launch(void* const* d_in, const int* in_sizes, int n_in,
                              void* d_out, int out_size, void* d_ws, size_t ws_size,
                              hipStream_t stream)
{
    (void)n_in; (void)out_size; (void)ws_size;
    const float* verts0 = (const float*)d_in[0];
    const int*   faces0 = (const int*)d_in[1];
    const float* f3     = (const float*)d_in[2];
    const float* f4     = (const float*)d_in[3];
    const float* f5     = (const float*)d_in[4];
    const float* cc     = (const float*)d_in[5];
    const float* cf     = (const float*)d_in[6];
    const int*   ims    = (const int*)d_in[7];
    float* out = (float*)d_out;

    const int N0 = in_sizes[0] / 3;
    const int F0 = in_sizes[1] / 3;
    const int twoF0 = 2 * F0;
    const int N1 = N0 + twoF0;
    const int F1 = 4 * F0;
    const int twoF1 = 2 * F1;
    const int N2 = N1 + twoF1;

    // workspace carve-out (16B aligned), < 1 MB total
    char* w = (char*)d_ws;
    auto carve = [&](size_t bytes) { void* p = (void*)w; w += (bytes + 15) & ~(size_t)15; return p; };
    float*    verts1 = (float*)   carve((size_t)N1 * 3 * 4);
    float*    verts2 = (float*)   carve((size_t)N2 * 3 * 4);
    unsigned* keys0  = (unsigned*)carve((size_t)twoF0 * 4);
    unsigned* uniqA  = (unsigned*)carve((size_t)twoF0 * 4);
    int*      invA   = (int*)     carve((size_t)twoF0 * 4);
    int*      UA     = (int*)     carve(16);
    int*      faces1 = (int*)     carve((size_t)F1 * 3 * 4);
    unsigned* keys1  = (unsigned*)carve((size_t)twoF1 * 4);
    unsigned* uniqB  = (unsigned*)carve((size_t)twoF1 * 4);
    int*      UB     = (int*)     carve(16);

    // ---- unpool stage A (from faces0) ----
    k_build_keys<<<ceil_div_i(twoF0, 256), 256, 0, stream>>>(faces0, F0, keys0);
    int PA = npow2_i(twoF0);
    k_sort_unique<<<1, 1024, (size_t)(PA + 1024) * 4, stream>>>(keys0, twoF0, PA, uniqA, UA);
    k_inverse<<<ceil_div_i(twoF0, 256), 256, 0, stream>>>(keys0, twoF0, uniqA, UA, invA);
    k_midverts<<<ceil_div_i(N1, 256), 256, 0, stream>>>(verts0, N0, uniqA, twoF0, verts1);
    k_faces1<<<ceil_div_i(F0, 256), 256, 0, stream>>>(faces0, F0, invA, N0, faces1);

    // ---- unpool stage B (from faces1; inverse not needed, faces2 is discarded) ----
    k_build_keys<<<ceil_div_i(twoF1, 256), 256, 0, stream>>>(faces1, F1, keys1);
    int PB = npow2_i(twoF1);
    k_sort_unique<<<1, 1024, (size_t)(PB + 1024) * 4, stream>>>(keys1, twoF1, PB, uniqB, UB);
    k_midverts<<<ceil_div_i(N2, 256), 256, 0, stream>>>(verts1, N1, uniqB, twoF1, verts2);

    // ---- perception pooling directly into output column blocks ----
    k_pool<<<N2, 256, 0, stream>>>(verts2, f3, f4, f5, cc, cf, ims, out, 0);     // cols 0:1280
    k_pool<<<N1, 256, 0, stream>>>(verts1, f3, f4, f5, cc, cf, ims, out, 1280);  // rows <N1
    k_pool<<<N0, 256, 0, stream>>>(verts0, f3, f4, f5, cc, cf, ims, out, 2560);  // rows <N0

    // ---- WMMA midpoint propagation (sources are strictly earlier rows, already written) ----
    const int cblk = CPOOL / 16; // 80
    {   // cols 1280:2560, rows [N1,N2) via uniqB
        int rb = twoF1 / 16, rem = twoF1 - rb * 16;
        int tiles = rb * cblk;
        if (tiles > 0)
            k_prop_wmma<<<ceil_div_i(tiles, 8), 256, 0, stream>>>(uniqB, tiles, out, 1280, N1);
        if (rem > 0)
            k_prop_tail<<<ceil_div_i(rem * CPOOL, 256), 256, 0, stream>>>(uniqB, rb * 16, rem, out, 1280, N1);
    }
    {   // cols 2560:3840, rows [N0,N1) via uniqA
        int rb = twoF0 / 16, rem = twoF0 - rb * 16;
        int tiles = rb * cblk;
        if (tiles > 0)
            k_prop_wmma<<<ceil_div_i(tiles, 8), 256, 0, stream>>>(uniqA, tiles, out, 2560, N0);
        if (rem > 0)
            k_prop_tail<<<ceil_div_i(rem * CPOOL, 256), 256, 0, stream>>>(uniqA, rb * 16, rem, out, 2560, N0);
    }
    {   // cols 2560:3840, rows [N1,N2) via uniqB (after previous pass)
        int rb = twoF1 / 16, rem = twoF1 - rb * 16;
        int tiles = rb * cblk;
        if (tiles > 0)
            k_prop_wmma<<<ceil_div_i(tiles, 8), 256, 0, stream>>>(uniqB, tiles, out, 2560, N1);
        if (rem > 0)
            k_prop_tail<<<ceil_div_i(rem * CPOOL, 256), 256, 0, stream>>>(uniqB, rb * 16, rem, out, 2560, N1);
    }

    // ---- vertex coordinate columns ----
    k_vertcols<<<ceil_div_i(N2, 256), 256, 0, stream>>>(verts2, N2, out);
}